// jslab_kt_21878563406227
// MI455X (gfx1250) — compile-verified
//
#include <hip/hip_runtime.h>

typedef __attribute__((ext_vector_type(2))) float v2f;
typedef __attribute__((ext_vector_type(8))) float v8f;

#define NF    1440          // forcing steps
#define NDTK  10            // basis knots
#define NSUB  60            // Euler substeps per forcing step
#define FDT   60.0f         // Euler dt (s)
#define FDTF  3600.0f       // forcing dt (s)
#define FDTK  518400.0f     // basis window (s)
#define CHUNK 45            // NF / 32 lanes

// ---------------------------------------------------------------------------
// Kernel A: Kt(1440x2) = rownorm(gauss(1440x10)) @ exp(pk)(10x2)
// One wave per 16-row tile, V_WMMA_F32_16X16X4_F32, K padded 10 -> 12.
// ---------------------------------------------------------------------------
__global__ __launch_bounds__(32) void basis_wmma(const float* __restrict__ pk,
                                                 float* __restrict__ Kt) {
  const int t    = blockIdx.x;      // 90 tiles of 16 rows
  const int lane = threadIdx.x;     // wave32
  const int half = lane >> 4;       // 0: lanes 0-15, 1: lanes 16-31
  const int col  = lane & 15;
  const int row  = t * 16 + (lane & 15);

  // normalized gaussian basis weights for this row
  float w[12];
  float s = 0.0f;
  const float gt = (float)row * FDTF;
  #pragma unroll
  for (int j = 0; j < NDTK; ++j) {
    float d = gt - (0.5f * FDTK + (float)j * FDTK);
    float e = expf(-0.5f * d * d / (FDTK * FDTK));
    w[j] = e;
    s += e;
  }
  w[10] = 0.0f; w[11] = 0.0f;
  const float inv_s = 1.0f / s;

  v8f c = {};
  #pragma unroll
  for (int kc = 0; kc < 3; ++kc) {
    const int kbase = kc * 4;
    // A 16x4 f32: lanes 0-15 -> v0:K+0, v1:K+1 ; lanes 16-31 -> v0:K+2, v1:K+3
    v2f a;
    a.x = w[kbase + (half ? 2 : 0)] * inv_s;
    a.y = w[kbase + (half ? 3 : 1)] * inv_s;
    // B 4x16 f32: v0 lanes0-15 row K+0, lanes16-31 row K+2;
    //             v1 lanes0-15 row K+1, lanes16-31 row K+3
    const int kb0 = kbase + (half ? 2 : 0);
    const int kb1 = kbase + (half ? 3 : 1);
    v2f b;
    b.x = (col < 2 && kb0 < NDTK) ? expf(pk[kb0 * 2 + col]) : 0.0f;
    b.y = (col < 2 && kb1 < NDTK) ? expf(pk[kb1 * 2 + col]) : 0.0f;
    c = __builtin_amdgcn_wmma_f32_16x16x4_f32(false, a, false, b,
                                              (short)0, c, false, false);
  }
  // C/D: VGPR i -> row M = i + 8*half, column = lane&15 ; keep cols 0,1
  if (col < 2) {
    #pragma unroll
    for (int i = 0; i < 8; ++i) {
      const int m = i + half * 8;
      Kt[(t * 16 + m) * 2 + col] = c[i];
    }
  }
}

// ---------------------------------------------------------------------------
// Kernel B: per forcing interval, compose the 60 substep affine maps.
//   w -> alpha*w + beta  (complex),  alpha = (1 - DT*K1) - i*DT*fc,
//   beta = DT*K0*(tax + i*tay).  Store (a, b) per interval.
// ---------------------------------------------------------------------------
__global__ void interval_maps(const float* __restrict__ Kt,
                              const float* __restrict__ TAx,
                              const float* __restrict__ TAy,
                              const float* __restrict__ fcp,
                              float4* __restrict__ maps) {
  const int iout = blockIdx.x * blockDim.x + threadIdx.x;
  if (iout >= NF) return;
  const float fc = fcp[0];
  const int sup = (iout + 1 < NF) ? iout + 1 : NF - 1;  // jax: itsup=-1 -> n-1
  const float tax0 = TAx[iout], tax1 = TAx[sup];
  const float tay0 = TAy[iout], tay1 = TAy[sup];
  const float ktb0 = Kt[sup * 2 + 0], ktb1 = Kt[sup * 2 + 1];

  float ax = 1.0f, ay = 0.0f, bx = 0.0f, by = 0.0f;  // identity map
  for (int iin = 0; iin < NSUB; ++iin) {
    // Kt[it-1] with jax semantics: -1 -> last row, >=NF clamps to last row
    int i1 = iout * NSUB + iin - 1;
    int ia = (i1 < 0) ? (NF - 1) : ((i1 >= NF) ? (NF - 1) : i1);
    const float aa = (float)iin / (float)NSUB;
    const float om = 1.0f - aa;
    const float k0  = om * Kt[ia * 2 + 0] + aa * ktb0;
    const float k1  = om * Kt[ia * 2 + 1] + aa * ktb1;
    const float tax = om * tax0 + aa * tax1;
    const float tay = om * tay0 + aa * tay1;

    const float arx = 1.0f - FDT * k1;   // Re(alpha)
    const float ary = -FDT * fc;         // Im(alpha)
    const float brx = FDT * k0 * tax;    // Re(beta)
    const float bry = FDT * k0 * tay;    // Im(beta)

    // compose step after current: a' = alpha*a ; b' = alpha*b + beta
    const float nax = arx * ax - ary * ay;
    const float nay = arx * ay + ary * ax;
    const float nbx = arx * bx - ary * by + brx;
    const float nby = arx * by + ary * bx + bry;
    ax = nax; ay = nay; bx = nbx; by = nby;
  }
  maps[iout] = make_float4(ax, ay, bx, by);
}

// ---------------------------------------------------------------------------
// Kernel C: one wave32. Lane L composes intervals [45L, 45L+45), a 5-step
// shuffle scan composes across lanes, then each lane replays its chunk
// emitting U[i], V[i] (state before interval i, starting from w0 = 0).
// ---------------------------------------------------------------------------
__global__ __launch_bounds__(32) void scan_emit(const float4* __restrict__ maps,
                                                float* __restrict__ U,
                                                float* __restrict__ V) {
  const int lane = threadIdx.x;
  const int base = lane * CHUNK;

  // compose this lane's chunk
  float ax = 1.0f, ay = 0.0f, bx = 0.0f, by = 0.0f;
  for (int j = 0; j < CHUNK; ++j) {
    const float4 m = maps[base + j];
    const float nax = m.x * ax - m.y * ay;
    const float nay = m.x * ay + m.y * ax;
    const float nbx = m.x * bx - m.y * by + m.z;
    const float nby = m.x * by + m.y * bx + m.w;
    ax = nax; ay = nay; bx = nbx; by = nby;
  }

  // inclusive Hillis-Steele scan across the wave (mine ∘ prev)
  for (int d = 1; d < 32; d <<= 1) {
    const float pax = __shfl_up(ax, d, 32);
    const float pay = __shfl_up(ay, d, 32);
    const float pbx = __shfl_up(bx, d, 32);
    const float pby = __shfl_up(by, d, 32);
    if (lane >= d) {
      const float nax = ax * pax - ay * pay;
      const float nay = ax * pay + ay * pax;
      const float nbx = ax * pbx - ay * pby + bx;
      const float nby = ax * pby + ay * pbx + by;
      ax = nax; ay = nay; bx = nbx; by = nby;
    }
  }

  // exclusive prefix for this lane
  float cax = __shfl_up(ax, 1, 32);
  float cay = __shfl_up(ay, 1, 32);
  float cbx = __shfl_up(bx, 1, 32);
  float cby = __shfl_up(by, 1, 32);
  if (lane == 0) { cax = 1.0f; cay = 0.0f; cbx = 0.0f; cby = 0.0f; }

  // replay: emit state (w0=0 -> state == B of accumulated map), then apply
  for (int j = 0; j < CHUNK; ++j) {
    const int i = base + j;
    U[i] = cbx;
    V[i] = cby;
    const float4 m = maps[i];
    const float nax = m.x * cax - m.y * cay;
    const float nay = m.x * cay + m.y * cax;
    const float nbx = m.x * cbx - m.y * cby + m.z;
    const float nby = m.x * cby + m.y * cbx + m.w;
    cax = nax; cay = nay; cbx = nbx; cby = nby;
  }
}

// ---------------------------------------------------------------------------
extern "C" void kernel_launch(void* const* d_in, const int* in_sizes, int n_in,
                              void* d_out, int out_size, void* d_ws, size_t ws_size,
                              hipStream_t stream) {
  const float* pk  = (const float*)d_in[0];
  const float* TAx = (const float*)d_in[1];
  const float* TAy = (const float*)d_in[2];
  const float* fc  = (const float*)d_in[3];
  float* out = (float*)d_out;                 // U[0..NF) then V[0..NF)

  float*  Kt   = (float*)d_ws;                                    // 2880 f32
  float4* maps = (float4*)((char*)d_ws + NF * 2 * sizeof(float)); // 1440 f4 (16B aligned: 11520 % 16 == 0)

  basis_wmma   <<<NF / 16, 32, 0, stream>>>(pk, Kt);
  interval_maps<<<(NF + 255) / 256, 256, 0, stream>>>(Kt, TAx, TAy, fc, maps);
  scan_emit    <<<1, 32, 0, stream>>>(maps, out, out + NF);
}